// TorchFactorExtractor_2138893713511
// MI455X (gfx1250) — compile-verified
//
#include <hip/hip_runtime.h>
#include <math.h>
#include <stdint.h>

#ifndef __has_builtin
#define __has_builtin(x) 0
#endif

#define TPB  256
#define TILE 256
#define NSH  316          // TILE + 60 staged elements per series
#define EPSF 1e-12f

constexpr int kL  = 1024;
constexpr int kLP = 964;   // kL - 60
constexpr int kNF = 158;   // 13 + 5*29

typedef float v2f __attribute__((ext_vector_type(2)));
typedef float v8f __attribute__((ext_vector_type(8)));

#if defined(__gfx1250__) && __has_builtin(__builtin_amdgcn_wmma_f32_16x16x4_f32)
#define USE_WMMA 1
#else
#define USE_WMMA 0
#endif

#if defined(__gfx1250__) && __has_builtin(__builtin_amdgcn_global_load_async_to_lds_b32) && __has_builtin(__builtin_amdgcn_s_wait_asynccnt)
#define USE_ASYNC 1
typedef __attribute__((address_space(1))) int as1_int;   // global
typedef __attribute__((address_space(3))) int as3_int;   // LDS
#else
#define USE_ASYNC 0
#endif

// Rolling sum of s[tl-w+1+shift .. tl+shift] for this lane's position tl = wb + lane.
// WMMA path: D = A(16x4, ones in rows 0 and 8) x B(4x16 Hankel of the data) + C,
// chained over ceil(w/4) K-chunks. Full chunks need no masking/clamping; only the
// single tail chunk (w%4 != 0) masks data in B via value-select (v_cndmask, not
// divergent loads). A stays all-ones so the sum is invariant to K-slot permutation.
__device__ __forceinline__ float rollsum(const float* __restrict__ s, int wb, int w,
                                         int lane, int shift) {
#if USE_WMMA
  const int n    = lane & 15;          // B/D column handled by this lane
  const int koff = (lane >> 4) << 1;   // lanes 16-31 supply K rows 2,3 of each chunk
  const float aw = ((lane & 7) == 0) ? 1.0f : 0.0f;  // A rows 0 and 8 = ones
  v2f a; a.x = aw; a.y = aw;
  const int nfull = w >> 2;            // chunks fully inside the window
  const int rem   = w & 3;
  float r0 = 0.0f, r1 = 0.0f;
  for (int g = 0; g < 2; ++g) {        // two 16-position groups per wave
    v8f acc = {0.0f,0.0f,0.0f,0.0f,0.0f,0.0f,0.0f,0.0f};
    const int nb = wb + (g << 4) + n - w + 1 + shift;
    for (int ck = 0; ck < nfull; ++ck) {
      const int i0 = nb + (ck << 2) + koff;   // always in [0, NSH-1]
      v2f bb; bb.x = s[i0]; bb.y = s[i0 + 1];
      acc = __builtin_amdgcn_wmma_f32_16x16x4_f32(false, a, false, bb, (short)0, acc,
                                                  false, false);
    }
    if (rem) {                          // uniform branch (w uniform across block)
      const int ka = (nfull << 2) + koff;
      int i0 = nb + ka;
      int i1 = i0 + 1;
      i0 = (i0 < 0) ? 0 : ((i0 > NSH - 1) ? NSH - 1 : i0);
      i1 = (i1 < 0) ? 0 : ((i1 > NSH - 1) ? NSH - 1 : i1);
      const float x0 = s[i0];           // unconditional loads ...
      const float x1 = s[i1];
      v2f bb;
      bb.x = (ka     < w) ? x0 : 0.0f;  // ... masked by value-select
      bb.y = (ka + 1 < w) ? x1 : 0.0f;
      acc = __builtin_amdgcn_wmma_f32_16x16x4_f32(false, a, false, bb, (short)0, acc,
                                                  false, false);
    }
    if (g == 0) r0 = acc[0]; else r1 = acc[0];
  }
  return (lane < 16) ? r0 : r1;  // lane<16: row m=0 of group0; lane>=16: row m=8 of group1
#else
  const int b0 = wb + lane - w + 1 + shift;
  float sum = 0.0f;
  for (int k = 0; k < w; ++k) sum += s[b0 + k];
  return sum;
#endif
}

__global__ __launch_bounds__(TPB)
void factor158_kernel(const float* __restrict__ x, float* __restrict__ out) {
  __shared__ float sC[NSH], sH[NSH], sL[NSH], sV[NSH];
  __shared__ float sLV[NSH], sRT[NSH], sRG[NSH], sWV[NSH];

  const int tid  = threadIdx.x;
  const int lane = tid & 31;
  const int b    = blockIdx.y;
  const int j0   = blockIdx.x * TILE;

  // ---- stage H/L/C/V tile into LDS (async global->LDS on CDNA5) ----
  for (int i = tid; i < NSH; i += TPB) {
    int g = j0 + i; g = (g > kL - 1) ? (kL - 1) : g;   // clamp only matters for invalid tail
    const float* p = x + ((size_t)(b * kL + g)) * 6;
#if USE_ASYNC
    __builtin_amdgcn_global_load_async_to_lds_b32((as1_int*)(p + 1), (as3_int*)&sH[i], 0, 0);
    __builtin_amdgcn_global_load_async_to_lds_b32((as1_int*)(p + 2), (as3_int*)&sL[i], 0, 0);
    __builtin_amdgcn_global_load_async_to_lds_b32((as1_int*)(p + 3), (as3_int*)&sC[i], 0, 0);
    __builtin_amdgcn_global_load_async_to_lds_b32((as1_int*)(p + 4), (as3_int*)&sV[i], 0, 0);
#else
    sH[i] = p[1]; sL[i] = p[2]; sC[i] = p[3]; sV[i] = p[4];
#endif
  }
#if USE_ASYNC
  __builtin_amdgcn_s_wait_asynccnt(0);
#endif
  __syncthreads();

  // ---- derived series: logv, ret1, lvchg, wv ----
  for (int i = tid; i < NSH; i += TPB) {
    const float vv = sV[i];
    sLV[i] = logf(vv + 1.0f);
    if (i < NSH - 1) {
      const float r = sC[i + 1] / sC[i];
      sRT[i] = r;
      sRG[i] = logf(sV[i + 1] / (vv + EPSF) + 1.0f);
      sWV[i] = fabsf(r - 1.0f) * sV[i + 1];
    }
  }
  __syncthreads();

  const int  j     = j0 + tid;
  const bool valid = (j < kLP);
  const int  tl    = 60 + tid;           // local index of output time t
  const int  wb    = 60 + (tid & ~31);   // local t of lane 0 of this wave

  const float c = sC[tl], h = sH[tl], l = sL[tl], v = sV[tl];
  int gt = j0 + tl; gt = (gt > kL - 1) ? (kL - 1) : gt;
  const float* pt = x + ((size_t)(b * kL + gt)) * 6;
  const float o  = pt[0];
  const float vw = pt[5];

  float* po = out + ((size_t)b * kLP + (valid ? j : 0)) * kNF;

  if (valid) {
    const float hl = h - l + EPSF;
    const float mo = fmaxf(o, c), mi = fminf(o, c);
    po[0]  = (c - o) / o;
    po[1]  = (h - l) / o;
    po[2]  = (c - o) / hl;
    po[3]  = (h - mo) / o;
    po[4]  = (h - mo) / hl;
    po[5]  = (mi - l) / o;
    po[6]  = (mi - l) / hl;
    po[7]  = (2.0f * c - h - l) / o;
    po[8]  = (2.0f * c - h - l) / hl;
    po[9]  = o / c;
    po[10] = h / c;
    po[11] = l / c;
    po[12] = vw / c;
  }

  const int WS[5] = {5, 10, 20, 30, 60};
#pragma unroll 1
  for (int wi = 0; wi < 5; ++wi) {
    const int   w   = WS[wi];
    const float fw  = (float)w;
    const float tm  = 0.5f * (fw - 1.0f);
    const float tvv = (fw * fw - 1.0f) / 12.0f;
    const int base  = tl - w + 1;   // raw-series windows [t-w+1 .. t]
    const int baseD = tl - w;       // diff-series windows [t-w .. t-1]

    // WMMA rolling sums (all threads, uniform EXEC)
    const float S1c  = rollsum(sC,  wb, w, lane, 0);
    const float S1v  = rollsum(sV,  wb, w, lane, 0);
    const float S1lv = rollsum(sLV, wb, w, lane, 0);
    const float S1rt = rollsum(sRT, wb, w, lane, -1);
    const float S1rg = rollsum(sRG, wb, w, lane, -1);
    const float S1wv = rollsum(sWV, wb, w, lane, -1);

    const float ym  = S1c  / fw;
    const float vmn = S1v  / fw;
    const float lvm = S1lv / fw;
    const float rtm = S1rt / fw;
    const float rgm = S1rg / fw;
    const float wvm = S1wv / fw;

    // regression / variance on close (two-pass like reference)
    float syy = 0.0f, sty = 0.0f;
    for (int k = 0; k < w; ++k) {
      const float d = sC[base + k] - ym;
      syy += d * d;
      sty += ((float)k - tm) * d;
    }
    const float cov   = sty / fw;
    const float yvar  = syy / fw;
    const float slope = cov / tvv;
    const float stdv  = sqrtf(yvar);
    const float rsqr  = cov * cov / (tvv * yvar + EPSF);
    const float resi  = c - ((ym - slope * tm) + slope * (fw - 1.0f));

    // max/min + first-occurrence arg
    float hmax = -3.4e38f, lmin = 3.4e38f;
    int ima = 0, imi = 0;
    for (int k = 0; k < w; ++k) {
      const float hh = sH[base + k];
      if (hh > hmax) { hmax = hh; ima = k; }
      const float ll = sL[base + k];
      if (ll < lmin) { lmin = ll; imi = k; }
    }
    const float fimax = (float)ima / fw;
    const float fimin = (float)imi / fw;

    // quantiles (linear interp, via rank counting) + rank feature
    const float p8 = 0.8f * (fw - 1.0f);
    const float p2 = 0.2f * (fw - 1.0f);
    const int   k8 = (int)p8;
    const int   k2 = (int)p2;
    const float f8 = p8 - (float)k8;
    const float f2 = p2 - (float)k2;
    float q8l = 0.0f, q8h = 0.0f, q2l = 0.0f, q2h = 0.0f;
    int rc = 0;
    for (int i2 = 0; i2 < w; ++i2) {
      const float xi = sC[base + i2];
      int r = 0;
      for (int jj = 0; jj < w; ++jj) {
        const float xj = sC[base + jj];
        r += ((xj < xi) || (xj == xi && jj < i2)) ? 1 : 0;
      }
      if (r == k8)     q8l = xi;
      if (r == k8 + 1) q8h = xi;
      if (r == k2)     q2l = xi;
      if (r == k2 + 1) q2h = xi;
      rc += (xi <= c) ? 1 : 0;
    }
    const float qtlu  = q8l + f8 * (q8h - q8l);
    const float qtld  = q2l + f2 * (q2h - q2l);
    const float rankv = (float)rc / fw;
    const float rsv   = (c - lmin) / (hmax - lmin + EPSF);

    // dclose stats
    float dp = 0.0f, dn = 0.0f, spp = 0.0f, snn = 0.0f, sab = 0.0f;
    for (int k = 0; k < w; ++k) {
      const float d = sC[baseD + k + 1] - sC[baseD + k];
      dp  += (d > 0.0f) ? 1.0f : 0.0f;
      dn  += (d < 0.0f) ? 1.0f : 0.0f;
      spp += fmaxf(d, 0.0f);
      snn += fmaxf(-d, 0.0f);
      sab += fabsf(d);
    }
    const float cntp = dp / fw, cntn = dn / fw;
    const float sump = spp / (sab + EPSF);
    const float sumn = snn / (sab + EPSF);

    // volume std (mean from WMMA)
    float svv = 0.0f;
    for (int k = 0; k < w; ++k) { const float d = sV[base + k] - vmn; svv += d * d; }
    const float vma  = vmn / (v + EPSF);
    const float vstd = sqrtf(svv / fw) / (v + EPSF);

    // corr(close, logv)
    float cnum = 0.0f, sbb = 0.0f;
    for (int k = 0; k < w; ++k) {
      const float am = sC[base + k]  - ym;
      const float bm = sLV[base + k] - lvm;
      cnum += am * bm;
      sbb  += bm * bm;
    }
    const float corr = cnum / (sqrtf(syy * sbb) + EPSF);

    // corr(ret1, lvchg)
    float dnum = 0.0f, sra = 0.0f, srb = 0.0f;
    for (int k = 0; k < w; ++k) {
      const float am = sRT[baseD + k] - rtm;
      const float bm = sRG[baseD + k] - rgm;
      dnum += am * bm; sra += am * am; srb += bm * bm;
    }
    const float cord = dnum / (sqrtf(sra * srb) + EPSF);

    // wv dispersion
    float sww = 0.0f;
    for (int k = 0; k < w; ++k) { const float d = sWV[baseD + k] - wvm; sww += d * d; }
    const float wvma = sqrtf(sww / fw) / (wvm + EPSF);

    // dvol stats
    float vpp = 0.0f, vnn = 0.0f, vab = 0.0f;
    for (int k = 0; k < w; ++k) {
      const float d = sV[baseD + k + 1] - sV[baseD + k];
      vpp += fmaxf(d, 0.0f);
      vnn += fmaxf(-d, 0.0f);
      vab += fabsf(d);
    }
    const float vsump = vpp / (vab + EPSF);
    const float vsumn = vnn / (vab + EPSF);

    if (valid) {
      float* fo = po + 13 + wi * 29;
      fo[0]  = sC[tl - w] / c;     // roc
      fo[1]  = ym / c;             // ma
      fo[2]  = stdv / c;           // std
      fo[3]  = slope / c;
      fo[4]  = rsqr;
      fo[5]  = resi / c;
      fo[6]  = hmax / c;           // mx
      fo[7]  = lmin / c;           // mn
      fo[8]  = qtlu / c;
      fo[9]  = qtld / c;
      fo[10] = rankv;
      fo[11] = rsv;
      fo[12] = fimax;
      fo[13] = fimin;
      fo[14] = fimax - fimin;
      fo[15] = corr;
      fo[16] = cord;
      fo[17] = cntp;
      fo[18] = cntn;
      fo[19] = cntp - cntn;
      fo[20] = sump;
      fo[21] = sumn;
      fo[22] = sump - sumn;
      fo[23] = vma;
      fo[24] = vstd;
      fo[25] = wvma;
      fo[26] = vsump;
      fo[27] = vsumn;
      fo[28] = vsump - vsumn;
    }
  }
}

extern "C" void kernel_launch(void* const* d_in, const int* in_sizes, int n_in,
                              void* d_out, int out_size, void* d_ws, size_t ws_size,
                              hipStream_t stream) {
  (void)n_in; (void)out_size; (void)d_ws; (void)ws_size;
  const float* x = (const float*)d_in[0];
  float* out = (float*)d_out;
  const int nb = in_sizes[0] / (kL * 6);   // batch = 128
  dim3 grid((kLP + TILE - 1) / TILE, nb, 1);
  factor158_kernel<<<grid, TPB, 0, stream>>>(x, out);
}